// ResidualBlock_22746146799804
// MI455X (gfx1250) — compile-verified
//
#include <hip/hip_runtime.h>

typedef __attribute__((ext_vector_type(2))) float v2f;
typedef __attribute__((ext_vector_type(8))) float v8f;

#define DIM 128

// ---------------------------------------------------------------- init ----
__global__ void init_kernel(float* __restrict__ agg, float* __restrict__ deg,
                            long long nd, int n) {
  long long tid = (long long)blockIdx.x * blockDim.x + threadIdx.x;
  if (tid < nd) agg[tid] = 0.0f;
  if (tid < n)  deg[tid] = 1.0f;   // self loop contributes 1
}

// -------------------------------------------------------------- degree ----
__global__ void deg_kernel(const int* __restrict__ dst, float* __restrict__ deg,
                           int e) {
  int tid = blockIdx.x * blockDim.x + threadIdx.x;
  if (tid < e) atomicAdd(&deg[dst[tid]], 1.0f);
}

// ------------------------------------------------ GEMM1: h = x @ W_gcn ----
// One block = 16 rows of x staged in LDS; 8 waves each own a 16-col tile.
// f32 WMMA 16x16x4, K loop over 128.
__global__ __launch_bounds__(256) void gemm_xw_kernel(
    const float* __restrict__ x, const float* __restrict__ W,
    float* __restrict__ h, int nrows) {
  __shared__ float As[16 * DIM];
  int row0 = blockIdx.x * 16;
  if (row0 >= nrows) return;
  int tid = threadIdx.x;
  for (int i = tid; i < 16 * DIM; i += 256)
    As[i] = x[(size_t)row0 * DIM + i];
  __syncthreads();

  int wave = tid >> 5;
  int lane = tid & 31;
  int half = lane >> 4;     // 0: K pair {k,k+1}, 1: K pair {k+2,k+3}
  int l    = lane & 15;
  int n0   = wave * 16;

  v8f c = {};
  for (int k = 0; k < DIM; k += 4) {
    int kk = k + half * 2;
    v2f a, b;
    a.x = As[l * DIM + kk];
    a.y = As[l * DIM + kk + 1];
    b.x = W[(size_t)kk * DIM + n0 + l];          // B[k][n] = W[k][n]
    b.y = W[(size_t)(kk + 1) * DIM + n0 + l];
    c = __builtin_amdgcn_wmma_f32_16x16x4_f32(false, a, false, b,
                                              (short)0, c, false, false);
  }
#pragma unroll
  for (int r = 0; r < 8; ++r) {
    int row = row0 + half * 8 + r;               // C layout: lanes16-31 -> M+8
    h[(size_t)row * DIM + n0 + l] = c[r];
  }
}

// ------------------------------------------- edge scatter (wave / edge) ----
// norm = rsqrt(deg[s]) * rsqrt(deg[d]); agg[d] += h[s] * norm
__global__ __launch_bounds__(256) void scatter_kernel(
    const float* __restrict__ h, const int* __restrict__ src,
    const int* __restrict__ dst, const float* __restrict__ deg,
    float* __restrict__ agg, int e) {
  int gw = blockIdx.x * 8 + (threadIdx.x >> 5);  // global wave id = edge id
  int lane = threadIdx.x & 31;
  if (gw >= e) return;
  int s = src[gw];
  int d = dst[gw];
  float norm = rsqrtf(deg[s]) * rsqrtf(deg[d]);
  const float4* hs = (const float4*)(h + (size_t)s * DIM);
  float4 v = hs[lane];                           // coalesced 512B per wave
  float* ad = agg + (size_t)d * DIM + lane * 4;
  atomicAdd(ad + 0, v.x * norm);
  atomicAdd(ad + 1, v.y * norm);
  atomicAdd(ad + 2, v.z * norm);
  atomicAdd(ad + 3, v.w * norm);
}

// ---------------------- epilogue: a2 = relu(agg + h/deg + b), in place ----
__global__ void post_kernel(float* __restrict__ agg, const float* __restrict__ h,
                            const float* __restrict__ deg,
                            const float* __restrict__ b, long long nd) {
  long long tid = (long long)blockIdx.x * blockDim.x + threadIdx.x;
  if (tid >= nd) return;
  int node = (int)(tid >> 7);
  int dcol = (int)(tid & (DIM - 1));
  float v = agg[tid] + h[tid] / deg[node] + b[dcol];  // self-loop: h * (1/deg)
  agg[tid] = fmaxf(v, 0.0f);
}

// ------------------- GEMM2: out = relu(a2 @ W_ln^T + x) (fused epilogue) ----
__global__ __launch_bounds__(256) void gemm_ln_kernel(
    const float* __restrict__ a2, const float* __restrict__ Wln,
    const float* __restrict__ x, float* __restrict__ out, int nrows) {
  __shared__ float As[16 * DIM];
  int row0 = blockIdx.x * 16;
  if (row0 >= nrows) return;
  int tid = threadIdx.x;
  for (int i = tid; i < 16 * DIM; i += 256)
    As[i] = a2[(size_t)row0 * DIM + i];
  __syncthreads();

  int wave = tid >> 5;
  int lane = tid & 31;
  int half = lane >> 4;
  int l    = lane & 15;
  int n0   = wave * 16;

  v8f c = {};
  for (int k = 0; k < DIM; k += 4) {
    int kk = k + half * 2;
    v2f a, b;
    a.x = As[l * DIM + kk];
    a.y = As[l * DIM + kk + 1];
    b.x = Wln[(size_t)(n0 + l) * DIM + kk];      // B[k][n] = W_ln[n][k]
    b.y = Wln[(size_t)(n0 + l) * DIM + kk + 1];
    c = __builtin_amdgcn_wmma_f32_16x16x4_f32(false, a, false, b,
                                              (short)0, c, false, false);
  }
#pragma unroll
  for (int r = 0; r < 8; ++r) {
    int row = row0 + half * 8 + r;
    size_t idx = (size_t)row * DIM + n0 + l;
    out[idx] = fmaxf(c[r] + x[idx], 0.0f);       // residual + relu
  }
}

// ------------------------------------------------------------- launch ----
extern "C" void kernel_launch(void* const* d_in, const int* in_sizes, int n_in,
                              void* d_out, int out_size, void* d_ws, size_t ws_size,
                              hipStream_t stream) {
  const float* x     = (const float*)d_in[0];
  const int*   edge  = (const int*)d_in[1];    // [2, E]
  const float* W_gcn = (const float*)d_in[2];
  const float* b_gcn = (const float*)d_in[3];
  const float* W_ln  = (const float*)d_in[4];
  float* out = (float*)d_out;

  int N = in_sizes[0] / DIM;                   // 100000
  int E = in_sizes[1] / 2;                     // 1600000
  const int* src = edge;
  const int* dst = edge + E;

  float* h   = (float*)d_ws;                   // [N, D]
  float* agg = h + (size_t)N * DIM;            // [N, D] (becomes relu input)
  float* deg = agg + (size_t)N * DIM;          // [N]

  long long nd = (long long)N * DIM;
  int mtiles = (N + 15) / 16;                  // 6250, exact

  init_kernel<<<(int)((nd + 255) / 256), 256, 0, stream>>>(agg, deg, nd, N);
  deg_kernel<<<(E + 255) / 256, 256, 0, stream>>>(dst, deg, E);
  gemm_xw_kernel<<<mtiles, 256, 0, stream>>>(x, W_gcn, h, N);
  scatter_kernel<<<(E + 7) / 8, 256, 0, stream>>>(h, src, dst, deg, agg, E);
  post_kernel<<<(int)((nd + 255) / 256), 256, 0, stream>>>(agg, h, deg, b_gcn, nd);
  gemm_ln_kernel<<<mtiles, 256, 0, stream>>>(agg, W_ln, x, out, N);
}